// Stability_89988154786262
// MI455X (gfx1250) — compile-verified
//
#include <hip/hip_runtime.h>
#include <stdint.h>

// Problem constants (from reference)
#define T_CYCLES   256
#define N_STREAMS  524288
#define THRESH     0.05f

// Tiling
#define WAVE          32                       // one wave per block (wave32)
#define EPT           4                        // streams per thread
#define BLK_STREAMS   (WAVE * EPT)             // 128 streams per block
#define TC            16                       // cycles per LDS chunk
#define NCHUNK        (T_CYCLES / TC)          // 16 chunks
#define CHUNK_INTS    (TC * BLK_STREAMS)       // 2048 ints = 8 KB
#define CHUNK_BYTES   (CHUNK_INTS * 4)

typedef __attribute__((ext_vector_type(4))) uint32_t sgpr4_t;
typedef __attribute__((ext_vector_type(8))) uint32_t sgpr8_t;

// Issue one 2D TDM tile load: tile_w elems wide (4B each) x tile_h rows,
// row stride = row_stride elems, destination = LDS byte offset lds_off.
// D# layout per CDNA5 ISA ch.8 (group0 128b + group1 256b, 2-group form).
__device__ __forceinline__ void tdm_load_tile_2d(uint64_t gaddr, uint32_t lds_off,
                                                 uint32_t tile_w, uint32_t tile_h,
                                                 uint32_t tensor_w, uint32_t tensor_h,
                                                 uint32_t row_stride) {
  sgpr4_t g0;
  g0[0] = 1u;                                              // count=1 (valid), user mode
  g0[1] = lds_off;                                         // lds_addr (bytes)
  g0[2] = (uint32_t)gaddr;                                 // global_addr[31:0]
  g0[3] = ((uint32_t)(gaddr >> 32) & 0x01FFFFFFu)          // global_addr[56:32]
        | (2u << 30);                                      // type = 2 ("image")
  sgpr8_t g1;
  g1[0] = (2u << 16);                                      // wg_mask=0, data_size=2 (4B)
  g1[1] = (tensor_w & 0xFFFFu) << 16;                      // tensor_dim0[15:0] @ bits63:48
  g1[2] = ((tensor_w >> 16) & 0xFFFFu)                     // tensor_dim0[31:16]
        | ((tensor_h & 0xFFFFu) << 16);                    // tensor_dim1[15:0]
  g1[3] = ((tensor_h >> 16) & 0xFFFFu)                     // tensor_dim1[31:16]
        | ((tile_w & 0xFFFFu) << 16);                      // tile_dim0
  g1[4] = (tile_h & 0xFFFFu);                              // tile_dim1 (tile_dim2 = 0)
  g1[5] = row_stride;                                      // tensor_dim0_stride[31:0]
  g1[6] = 0u;                                              // stride[47:32], dim1_stride lo
  g1[7] = 0u;                                              // dim1_stride hi
  // Non-temporal: stream 512MB without polluting the 192MB L2.
  asm volatile("tensor_load_to_lds %0, %1 th:TH_LOAD_NT" :: "s"(g0), "s"(g1) : "memory");
}

__global__ void __launch_bounds__(WAVE)
stability_kernel(const float* __restrict__ src, const int* __restrict__ bits,
                 float* __restrict__ out) {
  extern __shared__ int smem[];                 // 2 x CHUNK_BYTES, double buffered
  const int tid = threadIdx.x;
  const int n0  = blockIdx.x * BLK_STREAMS;     // first stream of this block
  const uint64_t gbase = (uint64_t)(uintptr_t)bits + (uint64_t)n0 * 4ull;
  const uint64_t chunk_stride = (uint64_t)TC * (uint64_t)N_STREAMS * 4ull;

  // Prime the double-buffer pipeline with chunks 0 and 1.
  tdm_load_tile_2d(gbase,                 0,           BLK_STREAMS, TC,
                   N_STREAMS, T_CYCLES, N_STREAMS);
  tdm_load_tile_2d(gbase + chunk_stride,  CHUNK_BYTES, BLK_STREAMS, TC,
                   N_STREAMS, T_CYCLES, N_STREAMS);

  // Per-stream constants: -1 - clip(src, -1, 1)   (SCALE == 1)
  const float4 s = ((const float4*)(src + n0))[tid];
  const float m0 = -1.0f - fminf(fmaxf(s.x, -1.0f), 1.0f);
  const float m1 = -1.0f - fminf(fmaxf(s.y, -1.0f), 1.0f);
  const float m2 = -1.0f - fminf(fmaxf(s.z, -1.0f), 1.0f);
  const float m3 = -1.0f - fminf(fmaxf(s.w, -1.0f), 1.0f);

  int   c0 = 0,   c1 = 0,   c2 = 0,   c3 = 0;    // running one-counts
  float l0 = 0.f, l1 = 0.f, l2 = 0.f, l3 = 0.f;  // last unstable cycle
  float tf = 0.f;                                 // cycle counter (1..256)

  for (int i = 0; i < NCHUNK; ++i) {
    // Loads complete in order: <=1 outstanding means chunk i has landed.
    if (i + 1 < NCHUNK) __builtin_amdgcn_s_wait_tensorcnt(1);
    else                __builtin_amdgcn_s_wait_tensorcnt(0);
    asm volatile("" ::: "memory");               // keep ds loads after the wait

    const int base = (i & 1) ? CHUNK_INTS : 0;
#pragma unroll
    for (int r = 0; r < TC; ++r) {
      tf += 1.0f;
      const float inv2 = __builtin_amdgcn_rcpf(0.5f * tf);   // 2 / t
      const int4 b = *(const int4*)(smem + base + r * BLK_STREAMS + EPT * tid);
      c0 += b.x; c1 += b.y; c2 += b.z; c3 += b.w;
      const float p0 = fmaf((float)c0, inv2, m0);            // 2c/t - 1 - src
      const float p1 = fmaf((float)c1, inv2, m1);
      const float p2 = fmaf((float)c2, inv2, m2);
      const float p3 = fmaf((float)c3, inv2, m3);
      l0 = (fabsf(p0) > THRESH) ? tf : l0;                   // v_cmp(|x|) + cndmask
      l1 = (fabsf(p1) > THRESH) ? tf : l1;
      l2 = (fabsf(p2) > THRESH) ? tf : l2;
      l3 = (fabsf(p3) > THRESH) ? tf : l3;
    }
    asm volatile("" ::: "memory");               // all reads of this buffer done

    if (i + 2 < NCHUNK)                          // refill the buffer just consumed
      tdm_load_tile_2d(gbase + (uint64_t)(i + 2) * chunk_stride,
                       (uint32_t)(base * 4), BLK_STREAMS, TC,
                       N_STREAMS, T_CYCLES, N_STREAMS);
  }

  // stability = 1 - clamp(last, 1, T) / T
  float4 r;
  r.x = 1.0f - fminf(fmaxf(l0, 1.0f), 256.0f) * (1.0f / 256.0f);
  r.y = 1.0f - fminf(fmaxf(l1, 1.0f), 256.0f) * (1.0f / 256.0f);
  r.z = 1.0f - fminf(fmaxf(l2, 1.0f), 256.0f) * (1.0f / 256.0f);
  r.w = 1.0f - fminf(fmaxf(l3, 1.0f), 256.0f) * (1.0f / 256.0f);
  ((float4*)(out + n0))[tid] = r;
}

extern "C" void kernel_launch(void* const* d_in, const int* in_sizes, int n_in,
                              void* d_out, int out_size, void* d_ws, size_t ws_size,
                              hipStream_t stream) {
  (void)in_sizes; (void)n_in; (void)out_size; (void)d_ws; (void)ws_size;
  const float* src  = (const float*)d_in[0];   // [N] float32
  const int*   bits = (const int*)d_in[1];     // [T, N] int32 {0,1}
  float*       out  = (float*)d_out;           // [N] float32

  dim3 grid(N_STREAMS / BLK_STREAMS);          // 4096 single-wave blocks
  dim3 block(WAVE);
  stability_kernel<<<grid, block, 2 * CHUNK_BYTES, stream>>>(src, bits, out);
}